// Max_73521250173295
// MI455X (gfx1250) — compile-verified
//
#include <hip/hip_runtime.h>

// ---------------------------------------------------------------------------
// Segmented one-hot argmax over ragged segments (first-max tie-break).
// Memory-bound: ~260MB traffic -> ~11us floor at 23.3 TB/s on MI455X.
// CDNA5 path: wave32 (one wave per segment), async global->LDS staging
// (global_load_async_to_lds_b32 + s_wait_asynccnt) when the toolchain
// exposes the gfx1250 builtins; clean coalesced fallback otherwise.
// ---------------------------------------------------------------------------

#if defined(__gfx1250__) &&                                                   \
    __has_builtin(__builtin_amdgcn_global_load_async_to_lds_b32) &&           \
    __has_builtin(__builtin_amdgcn_s_wait_asynccnt)
#define SEG_ASYNC 1
#else
#define SEG_ASYNC 0
#endif

#if SEG_ASYNC
typedef __attribute__((address_space(1))) int* gptr_i32;
typedef __attribute__((address_space(3))) int* lptr_i32;
#endif

#define SCAN_BLOCK 256
#define ITEMS_PER_THREAD 4
#define SEG_PER_CTA 8           // 256 threads = 8 wave32, one segment per wave
#define STAGE_CAP 224           // max segment size is ~193; 224 = 7*32

// ---- Stage 1: per-1024-chunk sums of the int64 size list -------------------
__global__ void __launch_bounds__(SCAN_BLOCK)
block_sums(const long long* __restrict__ sizes, unsigned* __restrict__ partial, int B) {
  __shared__ unsigned s[SCAN_BLOCK];
  const int t = threadIdx.x;
  const int base = blockIdx.x * (SCAN_BLOCK * ITEMS_PER_THREAD) + t * ITEMS_PER_THREAD;
  unsigned acc = 0;
  #pragma unroll
  for (int j = 0; j < ITEMS_PER_THREAD; ++j) {
    int i = base + j;
    if (i < B) acc += (unsigned)sizes[i];
  }
  s[t] = acc;
  __syncthreads();
  for (int d = SCAN_BLOCK / 2; d > 0; d >>= 1) {
    if (t < d) s[t] += s[t + d];
    __syncthreads();
  }
  if (t == 0) partial[blockIdx.x] = s[0];
}

// ---- Stage 2: tiny serial exclusive scan of the chunk sums -----------------
__global__ void scan_partials(const unsigned* __restrict__ partial,
                              unsigned* __restrict__ blockBase, int n) {
  unsigned acc = 0;
  for (int i = 0; i < n; ++i) { blockBase[i] = acc; acc += partial[i]; }
}

// ---- Stage 3: per-chunk exclusive scan -> per-segment start offsets --------
__global__ void __launch_bounds__(SCAN_BLOCK)
block_scan(const long long* __restrict__ sizes, const unsigned* __restrict__ blockBase,
           unsigned* __restrict__ offs, int B) {
  __shared__ unsigned s[SCAN_BLOCK];
  const int t = threadIdx.x;
  const int base = blockIdx.x * (SCAN_BLOCK * ITEMS_PER_THREAD) + t * ITEMS_PER_THREAD;
  unsigned v[ITEMS_PER_THREAD];
  unsigned tot = 0;
  #pragma unroll
  for (int j = 0; j < ITEMS_PER_THREAD; ++j) {
    int i = base + j;
    v[j] = (i < B) ? (unsigned)sizes[i] : 0u;
    tot += v[j];
  }
  s[t] = tot;
  __syncthreads();
  // Hillis-Steele inclusive scan over the 256 per-thread totals
  for (int d = 1; d < SCAN_BLOCK; d <<= 1) {
    unsigned add = (t >= d) ? s[t - d] : 0u;
    __syncthreads();
    s[t] += add;
    __syncthreads();
  }
  unsigned o = blockBase[blockIdx.x] + (s[t] - tot);   // exclusive within grid
  #pragma unroll
  for (int j = 0; j < ITEMS_PER_THREAD; ++j) {
    int i = base + j;
    if (i < B) offs[i] = o;
    o += v[j];
  }
}

// ---- Stage 4: one wave32 per segment: argmax + one-hot write ---------------
__global__ void __launch_bounds__(256)
seg_onehot_argmax(const float* __restrict__ x, const unsigned* __restrict__ offs,
                  const long long* __restrict__ sizes, float* __restrict__ out,
                  int nSeg) {
#if SEG_ASYNC
  __shared__ float stage[SEG_PER_CTA][STAGE_CAP];
#endif
  const int wv = threadIdx.x >> 5;
  const int lane = threadIdx.x & 31;
  const int seg = blockIdx.x * SEG_PER_CTA + wv;
  if (seg >= nSeg) return;

  const unsigned base = offs[seg];
  const int n = (int)sizes[seg];
  const float* xs = x + base;

  float best = -__builtin_huge_valf();
  int bi = 0x7fffffff;

#if SEG_ASYNC
  if (n <= STAGE_CAP) {
    float* sb = stage[wv];
    // Async DMA the segment into LDS: one b32 per lane per step,
    // 128B per wave-instruction, tracked on ASYNCcnt.
    for (int k = lane; k < n; k += 32) {
      __builtin_amdgcn_global_load_async_to_lds_b32(
          (gptr_i32)(xs + k),
          (lptr_i32)(sb + k),
          /*offset=*/0, /*cpol=*/0);
    }
    __builtin_amdgcn_s_wait_asynccnt(0);
    for (int k = lane; k < n; k += 32) {
      float v = sb[k];
      if (v > best) { best = v; bi = k; }   // strict '>' keeps first max per lane
    }
  } else
#endif
  {
    for (int k = lane; k < n; k += 32) {
      float v = xs[k];
      if (v > best) { best = v; bi = k; }
    }
  }

  // Wave32 reduction: max value, lowest global index on ties.
  #pragma unroll
  for (int d = 16; d > 0; d >>= 1) {
    float ov = __shfl_xor(best, d, 32);
    int   oi = __shfl_xor(bi,   d, 32);
    if (ov > best || (ov == best && oi < bi)) { best = ov; bi = oi; }
  }

  // Second pass writes every element of the segment exactly once
  // (segments tile [0,N) exactly -> no separate zero-fill needed).
  float* os = out + base;
  for (int k = lane; k < n; k += 32)
    os[k] = (k == bi) ? 1.0f : 0.0f;
}

// ---------------------------------------------------------------------------
extern "C" void kernel_launch(void* const* d_in, const int* in_sizes, int n_in,
                              void* d_out, int out_size, void* d_ws, size_t ws_size,
                              hipStream_t stream) {
  const float*     x     = (const float*)d_in[0];
  const long long* sizes = (const long long*)d_in[1];
  float*           out   = (float*)d_out;

  const int B = in_sizes[1];                       // number of segments
  const int chunk = SCAN_BLOCK * ITEMS_PER_THREAD; // 1024 sizes per scan block
  const int nb = (B + chunk - 1) / chunk;

  unsigned* offs      = (unsigned*)d_ws;           // [B]  segment start offsets
  unsigned* partial   = offs + B;                  // [nb] per-chunk sums
  unsigned* blockBase = partial + nb;              // [nb] exclusive chunk bases

  block_sums<<<nb, SCAN_BLOCK, 0, stream>>>(sizes, partial, B);
  scan_partials<<<1, 1, 0, stream>>>(partial, blockBase, nb);
  block_scan<<<nb, SCAN_BLOCK, 0, stream>>>(sizes, blockBase, offs, B);

  const int ctas = (B + SEG_PER_CTA - 1) / SEG_PER_CTA;
  seg_onehot_argmax<<<ctas, 256, 0, stream>>>(x, offs, sizes, out, B);
}